// Attention_53403623359087
// MI455X (gfx1250) — compile-verified
//
#include <hip/hip_runtime.h>
#include <hip/hip_bf16.h>
#include <math.h>

#define BZ 32
#define SRC_LEN 2048
#define DIM 1024
#define NCHUNK 8
#define ROWS_PER_CHUNK (SRC_LEN / NCHUNK) /* 256 */

typedef float v2f __attribute__((ext_vector_type(2)));
typedef float v8f __attribute__((ext_vector_type(8)));

// ---------------------------------------------------------------------------
// Generic row-major C = A @ B^T via V_WMMA_F32_16X16X4_F32 (wave32).
// A: M x K, B: N x K, C: M x N.  One 16x16 output tile per wave.
// A 16x4 f32 layout: VGPR0 = (M=lane&15, K=base+ (lane<16?0:2)), VGPR1 = K+1.
// B 4x16 f32 layout is symmetric (N = lane&15).  D: 8 VGPRs, std 16x16 layout.
// ---------------------------------------------------------------------------
__global__ void gemm_ABt_wmma(const float* __restrict__ A,
                              const float* __restrict__ B,
                              float* __restrict__ C,
                              int M, int N, int K) {
    const int wave = threadIdx.x >> 5;
    const int lane = threadIdx.x & 31;
    const int tile = blockIdx.x * (blockDim.x >> 5) + wave;
    const int mtiles = M >> 4;
    const int ntiles = N >> 4;
    if (tile >= mtiles * ntiles) return;   // wave-uniform: EXEC stays all-1s

    const int mt = tile % mtiles;
    const int nt = tile / mtiles;
    const int row = lane & 15;             // M-row for A, N-row for B
    const int kk  = (lane >> 4) << 1;      // 0 for lanes 0-15, 2 for lanes 16-31

    const float* arow = A + (size_t)(mt * 16 + row) * K;
    const float* brow = B + (size_t)(nt * 16 + row) * K;

    v8f c = {0.f, 0.f, 0.f, 0.f, 0.f, 0.f, 0.f, 0.f};
    for (int k = 0; k < K; k += 4) {
        v2f a = *(const v2f*)(arow + k + kk);   // A[m][k+kk], A[m][k+kk+1]
        v2f b = *(const v2f*)(brow + k + kk);   // B[n][k+kk], B[n][k+kk+1]
        c = __builtin_amdgcn_wmma_f32_16x16x4_f32(
                /*neg_a=*/false, a, /*neg_b=*/false, b,
                /*c_mod=*/(short)0, c, /*reuse_a=*/false, /*reuse_b=*/false);
    }

    // D layout: VGPR r -> (M = r + (lane<16?0:8), N = lane&15)
    const int mbase = mt * 16 + ((lane >> 4) << 3);
    const int ncol  = nt * 16 + row;
#pragma unroll
    for (int r = 0; r < 8; ++r) {
        C[(size_t)(mbase + r) * N + ncol] = c[r];
    }
}

// ---------------------------------------------------------------------------
// Fused scores + chunk-local softmax partials + partial context.
// grid = (NCHUNK, BZ), block = 256 threads (8 waves).
// Pass A: one row per wave (coalesced float4 reads), cross-lane reduce.
// Pass B: re-read chunk (L2-hot) to accumulate c_part = sum exp(s-m)*src_row.
// ---------------------------------------------------------------------------
__global__ void scores_partial_kernel(const float* __restrict__ src,
                                      const float* __restrict__ tgt_proj,
                                      const long long* __restrict__ lens,
                                      float* __restrict__ scores_raw,
                                      float* __restrict__ m_arr,
                                      float* __restrict__ z_arr,
                                      float* __restrict__ c_part) {
    const int chunk = blockIdx.x;
    const int b     = blockIdx.y;
    const int tid   = threadIdx.x;
    const int lane  = tid & 31;
    const int wave  = tid >> 5;

    const long long len = lens[b];
    const int row0 = chunk * ROWS_PER_CHUNK;

    float* cpart_out = c_part + ((size_t)b * NCHUNK + chunk) * DIM;
    float* sc_out    = scores_raw + (size_t)b * SRC_LEN + row0;

    if ((long long)row0 >= len) {
        // Fully masked chunk: emit -inf scores, zero partials, skip all reads.
        sc_out[tid] = -INFINITY;
        if (tid == 0) {
            m_arr[b * NCHUNK + chunk] = -INFINITY;
            z_arr[b * NCHUNK + chunk] = 0.0f;
        }
        float4 zero = {0.f, 0.f, 0.f, 0.f};
        ((float4*)cpart_out)[tid] = zero;
        return;
    }

    __shared__ __align__(16) float tp[DIM];
    __shared__ float ls[ROWS_PER_CHUNK];
    __shared__ float lp[ROWS_PER_CHUNK];
    __shared__ float red[256];

    // Stage tgt_proj[b] in LDS (4 KB).
    ((float4*)tp)[tid] = ((const float4*)(tgt_proj + (size_t)b * DIM))[tid];
    __syncthreads();

    const float* srcb = src + (size_t)b * SRC_LEN * DIM;

    // ---- Pass A: 256 scores, one row per wave per step ----
    for (int t = 0; t < ROWS_PER_CHUNK / 8; ++t) {
        const int rl = wave * (ROWS_PER_CHUNK / 8) + t;   // 0..255
        const int g  = row0 + rl;
        float s;
        if ((long long)g < len) {
            const float4* rowp = (const float4*)(srcb + (size_t)g * DIM);
            float4 acc = {0.f, 0.f, 0.f, 0.f};
#pragma unroll
            for (int i = 0; i < DIM / 128; ++i) {
                float4 v = rowp[lane + i * 32];
                float4 w = ((const float4*)tp)[lane + i * 32];
                acc.x += v.x * w.x; acc.y += v.y * w.y;
                acc.z += v.z * w.z; acc.w += v.w * w.w;
            }
            s = acc.x + acc.y + acc.z + acc.w;
#pragma unroll
            for (int off = 16; off > 0; off >>= 1)
                s += __shfl_xor(s, off, 32);
        } else {
            s = -INFINITY;
        }
        if (lane == 0) { ls[rl] = s; sc_out[rl] = s; }
    }
    __syncthreads();

    // ---- Chunk-local max ----
    red[tid] = ls[tid];
    __syncthreads();
#pragma unroll
    for (int sft = 128; sft > 0; sft >>= 1) {
        if (tid < sft) red[tid] = fmaxf(red[tid], red[tid + sft]);
        __syncthreads();
    }
    const float m = red[0];
    __syncthreads();

    // ---- p = exp(s - m), chunk-local sum ----
    float p = (ls[tid] == -INFINITY) ? 0.0f : expf(ls[tid] - m);
    lp[tid] = p;
    red[tid] = p;
    __syncthreads();
#pragma unroll
    for (int sft = 128; sft > 0; sft >>= 1) {
        if (tid < sft) red[tid] += red[tid + sft];
        __syncthreads();
    }
    if (tid == 0) {
        m_arr[b * NCHUNK + chunk] = m;
        z_arr[b * NCHUNK + chunk] = red[0];
    }

    // ---- Pass B: partial context, coalesced re-read (L2-resident chunk) ----
    const int rows_valid = (len - (long long)row0 >= ROWS_PER_CHUNK)
                               ? ROWS_PER_CHUNK : (int)(len - (long long)row0);
    const float4* srcb4 = (const float4*)(srcb + (size_t)row0 * DIM);
    float4 c4 = {0.f, 0.f, 0.f, 0.f};
    for (int i = 0; i < rows_valid; ++i) {
        if (i + 8 < rows_valid)
            __builtin_prefetch(&srcb4[(size_t)(i + 8) * (DIM / 4) + tid], 0, 1);
        const float pi = lp[i];
        float4 v = srcb4[(size_t)i * (DIM / 4) + tid];
        c4.x += pi * v.x; c4.y += pi * v.y;
        c4.z += pi * v.z; c4.w += pi * v.w;
    }
    ((float4*)cpart_out)[tid] = c4;
}

// ---------------------------------------------------------------------------
// Combine chunk partials: global softmax, align output, context, concat.
// grid = BZ, block = 256.
// ---------------------------------------------------------------------------
__global__ void combine_kernel(const float* __restrict__ tgt,
                               const float* __restrict__ scores_raw,
                               const float* __restrict__ m_arr,
                               const float* __restrict__ z_arr,
                               const float* __restrict__ c_part,
                               float* __restrict__ concat_ws,
                               float* __restrict__ out_align) {
    const int b   = blockIdx.x;
    const int tid = threadIdx.x;

    float mj[NCHUNK], wj[NCHUNK];
    float M = -INFINITY;
#pragma unroll
    for (int j = 0; j < NCHUNK; ++j) {
        mj[j] = m_arr[b * NCHUNK + j];
        M = fmaxf(M, mj[j]);
    }
    float Z = 0.0f;
#pragma unroll
    for (int j = 0; j < NCHUNK; ++j) {
        float w = (mj[j] == -INFINITY) ? 0.0f : expf(mj[j] - M);
        wj[j] = w;
        Z += z_arr[b * NCHUNK + j] * w;
    }
    const float invZ = 1.0f / Z;

    // align = exp(raw - M) / Z  (masked rows were stored as -inf -> 0)
    for (int s = tid; s < SRC_LEN; s += 256) {
        float raw = scores_raw[(size_t)b * SRC_LEN + s];
        out_align[(size_t)b * SRC_LEN + s] =
            (raw == -INFINITY) ? 0.0f : expf(raw - M) * invZ;
    }

    // context c = sum_j (w_j/Z) * c_part_j ; concat = [c, tgt]
    float4 c4 = {0.f, 0.f, 0.f, 0.f};
#pragma unroll
    for (int j = 0; j < NCHUNK; ++j) {
        const float w = wj[j] * invZ;
        float4 v = ((const float4*)(c_part + ((size_t)b * NCHUNK + j) * DIM))[tid];
        c4.x += w * v.x; c4.y += w * v.y;
        c4.z += w * v.z; c4.w += w * v.w;
    }
    ((float4*)(concat_ws + (size_t)b * 2 * DIM))[tid] = c4;
    ((float4*)(concat_ws + (size_t)b * 2 * DIM + DIM))[tid] =
        ((const float4*)(tgt + (size_t)b * DIM))[tid];
}

// ---------------------------------------------------------------------------
extern "C" void kernel_launch(void* const* d_in, const int* in_sizes, int n_in,
                              void* d_out, int out_size, void* d_ws, size_t ws_size,
                              hipStream_t stream) {
    const float*     src  = (const float*)d_in[0];      // (32, 2048, 1024) f32
    const float*     tgt  = (const float*)d_in[1];      // (32, 1, 1024)    f32
    const float*     Wg   = (const float*)d_in[2];      // (1024, 1024)     f32
    const float*     Wo   = (const float*)d_in[3];      // (1024, 2048)     f32
    const long long* lens = (const long long*)d_in[4];  // (32,)            i64

    float* attn_h = (float*)d_out;                 // 32*1024
    float* align  = (float*)d_out + BZ * DIM;      // 32*2048

    // Workspace layout (floats):
    float* ws         = (float*)d_ws;
    float* tgt_proj   = ws;                              // 32*1024
    float* scores_raw = tgt_proj + BZ * DIM;             // 32*2048
    float* m_arr      = scores_raw + BZ * SRC_LEN;       // 32*8
    float* z_arr      = m_arr + BZ * NCHUNK;             // 32*8
    float* c_part     = z_arr + BZ * NCHUNK;             // 32*8*1024
    float* concat_ws  = c_part + BZ * NCHUNK * DIM;      // 32*2048
    // total ~1.7 MB

    // 1) tgt_proj = tgt @ W_general^T  (WMMA f32: 2x64 = 128 tiles, 8 waves/blk)
    gemm_ABt_wmma<<<16, 256, 0, stream>>>(tgt, Wg, tgt_proj, BZ, DIM, DIM);

    // 2) fused scores + chunk softmax partials + partial context (streams src once)
    scores_partial_kernel<<<dim3(NCHUNK, BZ), 256, 0, stream>>>(
        src, tgt_proj, lens, scores_raw, m_arr, z_arr, c_part);

    // 3) combine partials -> align output + concat [c, tgt]
    combine_kernel<<<BZ, 256, 0, stream>>>(
        tgt, scores_raw, m_arr, z_arr, c_part, concat_ws, align);

    // 4) attn_h = concat @ W_out^T  (WMMA f32, K = 2048)
    gemm_ABt_wmma<<<16, 256, 0, stream>>>(concat_ws, Wo, attn_h, BZ, DIM, 2 * DIM);
}